// StaticTimingAnalysis_16561393894027
// MI455X (gfx1250) — compile-verified
//
#include <hip/hip_runtime.h>
#include <hip/hip_bf16.h>
#include <stdint.h>

// Problem constants (match reference)
#define L_LEVELS 32
#define P_PINS   131072
#define K_FANIN  2
#define N_NETS_C 1000000
#define TCLK     10.0f
#define NEG_INF  (-1e30f)

// Launch shape: persistent fused kernel, grid-wide software barrier.
#define NBLOCKS  256
#define NTHREADS 256
#define NTH      (NBLOCKS * NTHREADS)   // 65536 threads -> 2 pins/thread/level

// ---------------------------------------------------------------------------
// Native CDNA5 float-min memory atomic (no-return form, STOREcnt-tracked).
// Deterministic: min is order-independent and inputs are NaN-free.
// SCOPE_DEV: RMW resolves at L2 (device coherence point) -> correct across WGPs.
// ---------------------------------------------------------------------------
__device__ __forceinline__ void global_min_f32(float* addr, float val) {
    asm volatile("global_atomic_min_num_f32 %0, %1, off scope:SCOPE_DEV"
                 :: "v"(addr), "v"(val)
                 : "memory");
}

// ---------------------------------------------------------------------------
// Grid-wide barrier: atomic counter + generation word in workspace.
// Workspace is memset to 0 at the start of every kernel_launch (graph memset
// node), so state is fresh and deterministic on every replay.
// The release __threadfence() waits out all outstanding stores (incl. the
// inline-asm atomics); the acquire fence invalidates WGP$ so plain loads see
// other WGPs' writes.
// ---------------------------------------------------------------------------
__device__ __forceinline__ void grid_barrier(unsigned* cnt, unsigned* gen) {
    __syncthreads();
    if (threadIdx.x == 0) {
        unsigned g = __hip_atomic_load(gen, __ATOMIC_RELAXED, __HIP_MEMORY_SCOPE_AGENT);
        __threadfence();  // release: s_wait_storecnt/loadcnt + global_wb
        unsigned old = __hip_atomic_fetch_add(cnt, 1u, __ATOMIC_ACQ_REL,
                                              __HIP_MEMORY_SCOPE_AGENT);
        if (old == gridDim.x - 1u) {
            __hip_atomic_store(cnt, 0u, __ATOMIC_RELAXED, __HIP_MEMORY_SCOPE_AGENT);
            __hip_atomic_fetch_add(gen, 1u, __ATOMIC_RELEASE, __HIP_MEMORY_SCOPE_AGENT);
        } else {
            while (__hip_atomic_load(gen, __ATOMIC_RELAXED,
                                     __HIP_MEMORY_SCOPE_AGENT) == g) {
                __builtin_amdgcn_s_sleep(2);
            }
        }
        __builtin_amdgcn_fence(__ATOMIC_ACQUIRE, "agent");  // global_inv
    }
    __syncthreads();
}

// ---------------------------------------------------------------------------
// Fused STA: init + forward (gather-max) + backward (scatter-min via
// global_atomic_min_num_f32) + slack, with grid barriers between levels.
// ---------------------------------------------------------------------------
__global__ __launch_bounds__(NTHREADS)
void sta_fused_kernel(const float* __restrict__ pin_delays,    // [L*P]
                      const int*   __restrict__ src_idx,       // [L-1][P][K]
                      const int*   __restrict__ net_ids,       // [L-1][P][K]
                      const uint8_t* __restrict__ net_mask,    // [N_NETS] bool
                      float* __restrict__ arrival,             // [L*P]
                      float* __restrict__ required,             // [L*P]
                      float* __restrict__ slack,               // [L*P]
                      float* __restrict__ mask_out,            // [N_NETS]
                      unsigned* cnt, unsigned* gen)
{
    const int tid = blockIdx.x * blockDim.x + threadIdx.x;

    // ---- init: level-0 arrivals, required=TCLK everywhere, mask copy ----
    for (int p = tid; p < P_PINS; p += NTH)
        arrival[p] = pin_delays[p];
    for (int i = tid; i < L_LEVELS * P_PINS; i += NTH)
        required[i] = TCLK;
    for (int n = tid; n < N_NETS_C; n += NTH)
        mask_out[n] = net_mask[n] ? 1.0f : 0.0f;

    // ---- forward: arrival propagation (max over valid fanin) ----
    for (int l = 1; l < L_LEVELS; ++l) {
        grid_barrier(cnt, gen);   // level l-1 arrivals finalized, visible

        const float* __restrict__ arr_prev = arrival + (size_t)(l - 1) * P_PINS;
        const int*   __restrict__ s   = src_idx + (size_t)(l - 1) * P_PINS * K_FANIN;
        const int*   __restrict__ nid = net_ids + (size_t)(l - 1) * P_PINS * K_FANIN;
        const float* __restrict__ dl  = pin_delays + (size_t)l * P_PINS;
        float*       __restrict__ arr_cur = arrival + (size_t)l * P_PINS;

        for (int p = tid; p < P_PINS; p += NTH) {
            // gfx1250 prefetch of next level's arc data while gathers are in flight
            if (l + 1 < L_LEVELS) {
                __builtin_prefetch(s   + (size_t)P_PINS * K_FANIN + 2 * p, 0, 1);
                __builtin_prefetch(nid + (size_t)P_PINS * K_FANIN + 2 * p, 0, 1);
            }
            int2 sp = ((const int2*)s)[p];     // {src0, src1}
            int2 np = ((const int2*)nid)[p];   // {net0, net1}
            float d = dl[p];
            // branchless: issue both gathers unconditionally (both L2 requests
            // in flight), select with NEG_INF exactly like the reference.
            float a0 = arr_prev[sp.x] + d;
            float a1 = arr_prev[sp.y] + d;
            bool  v0 = net_mask[np.x] == 0;
            bool  v1 = net_mask[np.y] == 0;
            float a  = fmaxf(fmaxf(v0 ? a0 : NEG_INF, v1 ? a1 : NEG_INF), d);
            arr_cur[p] = a;
        }
    }

    // ---- backward: required propagation (scatter-min over valid fanout) ----
    // required[L-1] stays at TCLK (already initialized).
    for (int l = L_LEVELS - 2; l >= 0; --l) {
        grid_barrier(cnt, gen);   // required[l+1] finalized, visible

        const float* __restrict__ req_next = required + (size_t)(l + 1) * P_PINS;
        const float* __restrict__ dl_next  = pin_delays + (size_t)(l + 1) * P_PINS;
        const int*   __restrict__ s   = src_idx + (size_t)l * P_PINS * K_FANIN;
        const int*   __restrict__ nid = net_ids + (size_t)l * P_PINS * K_FANIN;
        float*       __restrict__ req_cur = required + (size_t)l * P_PINS;

        for (int p = tid; p < P_PINS; p += NTH) {
            float c = req_next[p] - dl_next[p];
            int2 sp = ((const int2*)s)[p];
            int2 np = ((const int2*)nid)[p];
            if (net_mask[np.x] == 0) global_min_f32(req_cur + sp.x, c);
            if (net_mask[np.y] == 0) global_min_f32(req_cur + sp.y, c);
        }
    }

    // ---- slack ----
    grid_barrier(cnt, gen);
    for (int i = tid; i < L_LEVELS * P_PINS; i += NTH)
        slack[i] = required[i] - arrival[i];
}

// ---------------------------------------------------------------------------
extern "C" void kernel_launch(void* const* d_in, const int* in_sizes, int n_in,
                              void* d_out, int out_size, void* d_ws, size_t ws_size,
                              hipStream_t stream) {
    (void)in_sizes; (void)n_in; (void)out_size;

    const float*   pin_delays = (const float*)d_in[0];
    const int*     src_idx    = (const int*)d_in[1];
    const int*     net_ids    = (const int*)d_in[2];
    const uint8_t* net_mask   = (const uint8_t*)d_in[3];  // jnp bool -> 1 byte

    const size_t LP = (size_t)L_LEVELS * P_PINS;
    float* out      = (float*)d_out;
    float* arrival  = out;               // [L*P]
    float* required = out + LP;          // [L*P]
    float* slack    = out + 2 * LP;      // [L*P]
    float* mask_out = out + 3 * LP;      // [N_NETS]

    // barrier state in workspace, zeroed every call (graph-capturable memset)
    unsigned* cnt = (unsigned*)d_ws;
    unsigned* gen = (unsigned*)((char*)d_ws + 128);
    size_t clr = ws_size < 256 ? ws_size : (size_t)256;
    hipMemsetAsync(d_ws, 0, clr, stream);

    sta_fused_kernel<<<dim3(NBLOCKS), dim3(NTHREADS), 0, stream>>>(
        pin_delays, src_idx, net_ids, net_mask,
        arrival, required, slack, mask_out, cnt, gen);
}